// EPGSimulation_12781822672903
// MI455X (gfx1250) — compile-verified
//
#include <hip/hip_runtime.h>

#define N_PULSES 64
#define BATCH    16384
#define N_STATES 21
#define TPB      64                       // 2 waves per block (wave32)
#define DW_PER_T (N_STATES * 5)           // 105 dwords per thread per pulse
#define DW_TILE  (TPB * DW_PER_T)         // 6720 dwords = 26880 B per tile
#define TILE_BYTES (DW_TILE * 4)

typedef unsigned int v4u __attribute__((ext_vector_type(4)));
typedef int          v4i __attribute__((ext_vector_type(4)));
typedef int          v8i __attribute__((ext_vector_type(8)));

__global__ __launch_bounds__(TPB) void epg_tdm_kernel(
    const float* __restrict__ fa,  const float* __restrict__ ph,
    const float* __restrict__ T1,  const float* __restrict__ T2,
    const float* __restrict__ B0,  const float* __restrict__ B1,
    const int*  __restrict__ TRp,
    float* __restrict__ out)
{
  __shared__ float smem[2 * DW_TILE];     // double-buffered output staging

  const int tid = threadIdx.x;
  const int b   = blockIdx.x * TPB + tid;

  // Per-thread constants (relaxation + off-resonance phase)
  const float TRf = (float)TRp[0];
  const float t1 = T1[b], t2 = T2[b], b0 = B0[b], b1 = B1[b];
  const float E1 = __expf(-TRf / t1);
  const float E2 = __expf(-TRf / t2);
  const float phi = 6.28318530717958647692f * b0 * TRf * 0.001f;
  float b0s, b0c; __sincosf(phi, &b0s, &b0c);
  const float epR = E2 * b0c, epI = E2 * b0s;   // combined E2 * exp(i*phi)
  const float rz  = 1.0f - E1;

  // Register-resident EPG state: 105 floats
  float FpR[N_STATES], FpI[N_STATES], FmR[N_STATES], FmI[N_STATES], Z[N_STATES];
  #pragma unroll
  for (int k = 0; k < N_STATES; ++k) { FpR[k]=0.f; FpI[k]=0.f; FmR[k]=0.f; FmI[k]=0.f; Z[k]=0.f; }
  Z[0] = 1.0f;

  #pragma unroll 1
  for (int p = 0; p < N_PULSES; ++p) {
    // Per-pulse RF coefficients
    const float a   = b1 * fa[p];
    const float phs = ph[p];
    float sh, chh; __sincosf(0.5f * a, &sh, &chh);
    const float ca2 = chh * chh, sa2 = sh * sh;
    const float sa  = 2.0f * sh * chh;
    const float cos_a = ca2 - sa2;
    float eI, eR; __sincosf(phs, &eI, &eR);
    const float e2R = eR * eR - eI * eI;
    const float e2I = 2.0f * eR * eI;
    const float t2R = e2R * sa2, t2I = e2I * sa2;
    const float saeI = sa * eI,  saeR = sa * eR;

    // Relax -> dephase -> mix -> shift (in place, 1-element delay for Fp)
    float pFpR = 0.f, pFpI = 0.f;
    #pragma unroll
    for (int k = 0; k < N_STATES; ++k) {
      float r = FpR[k], i = FpI[k];
      const float oFpR = r * epR - i * epI;
      const float oFpI = r * epI + i * epR;
      r = FmR[k]; i = FmI[k];
      const float oFmR =  r * epR + i * epI;
      const float oFmI = -r * epI + i * epR;
      float oZ = Z[k] * E1;
      if (k == 0) oZ += rz;

      const float FpnR = ca2*oFpR + t2R*oFmR - t2I*oFmI + saeI*oZ;
      const float FpnI = ca2*oFpI + t2R*oFmI + t2I*oFmR - saeR*oZ;
      const float FmnR = t2R*oFpR + t2I*oFpI + ca2*oFmR + saeI*oZ;
      const float FmnI = t2R*oFpI - t2I*oFpR + ca2*oFmI + saeR*oZ;
      const float Zn   = 0.5f*sa*(eR*(oFpI - oFmI) - eI*(oFpR + oFmR)) + cos_a*oZ;

      Z[k] = Zn;
      if (k >= 1) { FpR[k] = pFpR; FpI[k] = pFpI; FmR[k-1] = FmnR; FmI[k-1] = FmnI; }
      if (k == 1) { FpR[0] = FmnR; FpI[0] = -FmnI; }   // Fp_s[0] = conj(Fm_n[1])
      pFpR = FpnR; pFpI = FpnI;
    }
    FmR[N_STATES-1] = 0.f; FmI[N_STATES-1] = 0.f;

    const int buf = p & 1;

    // Gate LDS-buffer reuse on the TDM store issued 2 pulses ago for this buffer.
#if defined(__HIP_DEVICE_COMPILE__)
    if (__builtin_amdgcn_readfirstlane((unsigned)threadIdx.x >> 5) == 0u)
      __builtin_amdgcn_s_wait_tensorcnt(1);
#endif
    __syncthreads();

    // Stage this pulse's tile in LDS, exactly in global layout [b][state][5]
    float* dst = &smem[buf * DW_TILE + tid * DW_PER_T];
    #pragma unroll
    for (int k = 0; k < N_STATES; ++k) {
      dst[5*k + 0] = FpR[k];
      dst[5*k + 1] = FpI[k];
      dst[5*k + 2] = FmR[k];
      dst[5*k + 3] = FmI[k];
      dst[5*k + 4] = Z[k];
    }
    __syncthreads();   // all DS writes visible before the TDM reads LDS

#if defined(__HIP_DEVICE_COMPILE__)
    // One wave per workgroup fires the Tensor Data Mover: contiguous 26,880 B
    // LDS -> global DMA, fully coalesced, tracked by TENSORcnt.
    if (__builtin_amdgcn_readfirstlane((unsigned)threadIdx.x >> 5) == 0u) {
      const unsigned lds_addr =
          (unsigned)(unsigned long long)(void*)&smem[buf * DW_TILE];
      const unsigned long long gaddr = (unsigned long long)(void*)(out
          + ((unsigned long long)p * BATCH + (unsigned long long)blockIdx.x * TPB)
            * DW_PER_T);

      // D# group 0: count=1, lds_addr, global_addr[56:0], type=2
      v4u g0;
      g0[0] = 1u;
      g0[1] = lds_addr;
      g0[2] = (unsigned)gaddr;
      g0[3] = ((unsigned)(gaddr >> 32) & 0x1FFFFFFu) | (2u << 30);

      // D# group 1: mask=0, data_size=2 (4B), 1-D tensor/tile of DW_TILE dwords
      v8i g1;
      g1[0] = (int)(2u << 16);                            // data_size = 4 bytes
      g1[1] = (int)(((unsigned)DW_TILE & 0xFFFFu) << 16); // tensor_dim0[15:0] @ bits 63:48
      g1[2] = (int)((unsigned)DW_TILE >> 16);             // tensor_dim0[31:16]
      g1[3] = (int)(((unsigned)DW_TILE & 0xFFFFu) << 16); // tile_dim0 @ bits 127:112
      g1[4] = 0;                                          // tile_dim1 = tile_dim2 = 0
      g1[5] = (int)DW_TILE;                               // tensor_dim0_stride lo
      g1[6] = 0;
      g1[7] = 0;

      v4i g2 = {0, 0, 0, 0};
      v4i g3 = {0, 0, 0, 0};
#if __clang_major__ >= 23
      v8i g4 = {0, 0, 0, 0, 0, 0, 0, 0};
      __builtin_amdgcn_tensor_store_from_lds(g0, g1, g2, g3, g4, 0);
#else
      __builtin_amdgcn_tensor_store_from_lds(g0, g1, g2, g3, 0);
#endif
    }
#endif
  }
  // S_ENDPGM performs an implicit wait-idle, draining outstanding TENSORcnt.
}

extern "C" void kernel_launch(void* const* d_in, const int* in_sizes, int n_in,
                              void* d_out, int out_size, void* d_ws, size_t ws_size,
                              hipStream_t stream) {
  const float* fa = (const float*)d_in[0];
  const float* ph = (const float*)d_in[1];
  const float* T1 = (const float*)d_in[2];
  const float* T2 = (const float*)d_in[3];
  const float* B0 = (const float*)d_in[4];
  const float* B1 = (const float*)d_in[5];
  const int*   TR = (const int*)d_in[6];
  float* out = (float*)d_out;

  dim3 grid(BATCH / TPB), block(TPB);
  hipLaunchKernelGGL(epg_tdm_kernel, grid, block, 0, stream,
                     fa, ph, T1, T2, B0, B1, TR, out);
}